// LiquidNeuron_14345190769015
// MI455X (gfx1250) — compile-verified
//
#include <hip/hip_runtime.h>
#include <cmath>

// CDNA5 WMMA operand vector types
typedef __attribute__((ext_vector_type(2))) float v2f;
typedef __attribute__((ext_vector_type(8))) float v8f;

#define HDIM 8192
#define WAVES_PER_BLOCK 4
#define BLOCK (WAVES_PER_BLOCK * 32)

// Fused liquid-neuron RHS + STDP update.
//   out[0 .. H)            : dhdt
//   out[H .. H + H*H)      : W_new
// Each wave owns 16 columns j0..j0+15 and walks all rows in K=4 chunks.
// WMMA computes y[j] = sum_i h[i]*W[i][j] with A = h-chunk replicated over
// all 16 rows (so every accumulator row equals y). The W tile loaded for the
// B operand is reused in-register for the antisymmetric plasticity update.
__global__ __launch_bounds__(BLOCK) void liquid_fused_kernel(
    const float* __restrict__ h,
    const float* __restrict__ W,
    const float* __restrict__ tau,
    const float* __restrict__ eta,
    const float* __restrict__ lp,
    float* __restrict__ out)
{
    __shared__ float sh[HDIM];   // h      (reused H times per block -> LDS)
    __shared__ float sp[HDIM];   // last_pre

    const int tid = threadIdx.x;
    for (int idx = tid; idx < HDIM; idx += BLOCK) {
        sh[idx] = h[idx];
        sp[idx] = lp[idx];
    }
    __syncthreads();

    const int wave = tid >> 5;
    const int lane = tid & 31;
    const int n    = lane & 15;   // column within the 16-wide stripe
    const int kh   = lane >> 4;   // which K-half (K=0,1 vs K=2,3) this lane holds

    const int j = (blockIdx.x * WAVES_PER_BLOCK + wave) * 16 + n;

    const float e   = eta[0];
    const float hj  = sh[j];      // h[j]        (per-lane constant)
    const float lpj = sp[j];      // last_pre[j] (per-lane constant)

    const float2* sh2 = (const float2*)sh;  // packed b64 LDS reads
    const float2* sp2 = (const float2*)sp;

    float* __restrict__ outW = out + (size_t)HDIM;

    v8f acc = {};  // 16x16 f32 accumulator; all rows replicate y

    #pragma unroll 4
    for (int k0 = 0; k0 < HDIM; k0 += 4) {
        const int    i0 = k0 + kh * 2;              // even row index
        const size_t r0 = (size_t)i0 * HDIM + j;

        // A operand (16x4, rows replicated with h[k0..k0+3])
        const float2 hh = sh2[i0 >> 1];
        v2f a; a.x = hh.x; a.y = hh.y;

        // B operand (4x16 tile of W) -- streamed, read exactly once: NT load
        v2f b;
        b.x = __builtin_nontemporal_load(&W[r0]);
        b.y = __builtin_nontemporal_load(&W[r0 + HDIM]);

        // acc = A*B + acc  (v_wmma_f32_16x16x4_f32)
        acc = __builtin_amdgcn_wmma_f32_16x16x4_f32(
            /*neg_a=*/false, a, /*neg_b=*/false, b,
            /*c_mod=*/(short)0, acc, /*reuse_a=*/false, /*reuse_b=*/false);

        // STDP: W_new[i][j] = W[i][j] + eta*(h[i]*lp[j] - lp[i]*h[j])
        const float2 pp  = sp2[i0 >> 1];
        const float  wn0 = b.x + e * (hh.x * lpj - pp.x * hj);
        const float  wn1 = b.y + e * (hh.y * lpj - pp.y * hj);
        __builtin_nontemporal_store(wn0, &outW[r0]);          // written once: NT
        __builtin_nontemporal_store(wn1, &outW[r0 + HDIM]);
    }

    // Every accumulator row equals y (A rows replicated); lane n of the low
    // half holds y[j0+n] in acc[0]. Lanes 0-15 finish dhdt.
    if (kh == 0) {
        const float y = acc[0];
        out[j] = (-hj + tanhf(y)) / fabsf(tau[j]);
    }
}

extern "C" void kernel_launch(void* const* d_in, const int* in_sizes, int n_in,
                              void* d_out, int out_size, void* d_ws, size_t ws_size,
                              hipStream_t stream) {
    // setup_inputs order: t, h, W, tau, eta, last_pre
    const float* h   = (const float*)d_in[1];
    const float* W   = (const float*)d_in[2];
    const float* tau = (const float*)d_in[3];
    const float* eta = (const float*)d_in[4];
    const float* lp  = (const float*)d_in[5];
    float* out = (float*)d_out;

    const int stripes = HDIM / 16;                    // 512 column stripes
    const int grid    = stripes / WAVES_PER_BLOCK;    // 128 blocks x 128 threads
    liquid_fused_kernel<<<grid, BLOCK, 0, stream>>>(h, W, tau, eta, lp, out);
}